// BEVTrans_36644660969637
// MI455X (gfx1250) — compile-verified
//
#include <hip/hip_runtime.h>

typedef __attribute__((ext_vector_type(2))) float v2f;
typedef __attribute__((ext_vector_type(8))) float v8f;

#define NB      4
#define CH      256
#define IMG_H   96
#define IMG_W   320
#define HW      (IMG_H * IMG_W)        // 30720
#define W_RES   100
#define H_RES   5
#define D_RES   100
#define QTOT    (D_RES * W_RES)        // 10000 (q = d*100 + w)
#define NPTS    (H_RES * QTOT)         // 50000 per batch
#define NCHUNKS ((NB * NPTS) / 16)     // 12500 16-point WMMA chunks

// ---------------------------------------------------------------------------
// Kernel 1: project points with V_WMMA_F32_16X16X4_F32 (3x3 calib GEMM padded
// to 16x4 * 4x16) and emit per-point bilinear params: 4 clamped corner
// offsets + 4 validity-folded weights (32B AoS record per point).
// Param layout index: pidx = ((b*5 + h)*10000 + d*100 + w)
// ---------------------------------------------------------------------------
__global__ void bev_project_wmma(const float* __restrict__ calib,
                                 unsigned int* __restrict__ params) {
    const int lane  = threadIdx.x & 31;
    const int wave  = threadIdx.x >> 5;
    const int chunk = blockIdx.x * 8 + wave;   // blockDim.x == 256 -> 8 waves
    if (chunk >= NCHUNKS) return;              // wave-uniform guard (EXEC stays full)

    const int n = lane & 15;                   // point-in-chunk (lanes L and L+16 share n)
    const int g = chunk * 16 + n;              // global point id in (b,h,d,w) order
    const int b = g / NPTS;
    int r = g - b * NPTS;
    const int h = r / QTOT;  r -= h * QTOT;
    const int d = r / W_RES;
    const int w = r - d * W_RES;

    const float wx = -9.9f + 0.2f * (float)w;  // _span(-10,10,100)
    const float hy = -3.0f + 1.0f * (float)h;  // _span(-3.5,1.5,5)
    const float dz =  0.1f + 0.2f * (float)d;  // _span(0,20,100)

    const float* cb = calib + b * 9;           // wave-uniform (chunks don't cross batches)

    float img0, img1, img2;
#if __has_builtin(__builtin_amdgcn_wmma_f32_16x16x4_f32)
    // A operand 16x4 (M=i rows of calib, K=j): VGPR0 = {K0 | K2}, VGPR1 = {K1 | K3}
    float a0 = 0.0f, a1 = 0.0f;
    if (lane < 16) {
        if (lane < 3) { a0 = cb[lane * 3 + 0]; a1 = cb[lane * 3 + 1]; }
    } else {
        const int m = lane - 16;
        if (m < 3)    { a0 = cb[m * 3 + 2]; }  // K=2 ; K=3 column is zero
    }
    // B operand 4x16 (K=j rows of position, N=point): VGPR0 = {K0 | K2}, VGPR1 = {K1 | K3}
    float b0, b1;
    if (lane < 16) { b0 = wx; b1 = hy; }       // rows K=0, K=1
    else           { b0 = dz; b1 = 0.0f; }     // rows K=2, K=3(=0)

    v2f A; A.x = a0; A.y = a1;
    v2f B; B.x = b0; B.y = b1;
    v8f C = {};
    v8f D = __builtin_amdgcn_wmma_f32_16x16x4_f32(false, A, false, B,
                                                  (short)0, C, false, false);
    img0 = D[0];  img1 = D[1];  img2 = D[2];   // rows 0..2 for lanes 0..15
#else
    img0 = cb[0] * wx + cb[1] * hy + cb[2] * dz;
    img1 = cb[3] * wx + cb[4] * hy + cb[5] * dz;
    img2 = cb[6] * wx + cb[7] * hy + cb[8] * dz;
#endif

    if (lane < 16) {
        const float inv = 1.0f / img2;                  // img2 = dz > 0 (calib row2 = [0,0,1])
        const float ix = img0 * inv * (float)(IMG_W - 1);  // == ((2(u-0.5))+1)*0.5*(W-1)
        const float iy = img1 * inv * (float)(IMG_H - 1);
        const float fx0 = floorf(ix), fy0 = floorf(iy);
        const int ix0 = (int)fx0, iy0 = (int)fy0;
        const float fx = ix - fx0, fy = iy - fy0;

        unsigned int rec[8];
#pragma unroll
        for (int dy = 0; dy < 2; ++dy) {
#pragma unroll
            for (int dx = 0; dx < 2; ++dx) {
                const int xi = ix0 + dx, yi = iy0 + dy;
                float wt = (dx ? fx : 1.0f - fx) * (dy ? fy : 1.0f - fy);
                const bool valid = (xi >= 0) & (xi <= IMG_W - 1) &
                                   (yi >= 0) & (yi <= IMG_H - 1);
                if (!valid) wt = 0.0f;
                const int xic = min(max(xi, 0), IMG_W - 1);
                const int yic = min(max(yi, 0), IMG_H - 1);
                const int k = dy * 2 + dx;
                rec[k]     = (unsigned int)(yic * IMG_W + xic);
                rec[4 + k] = __float_as_uint(wt);
            }
        }
        uint4* dst = (uint4*)(params + (size_t)g * 8);
        dst[0] = make_uint4(rec[0], rec[1], rec[2], rec[3]);
        dst[1] = make_uint4(rec[4], rec[5], rec[6], rec[7]);
    }
}

// ---------------------------------------------------------------------------
// Kernel 2: gather + blend. Block = (tile of 128 q, h, b). Each thread owns
// one point (params held in registers for the whole loop) and sweeps 128
// channels; stores are coalesced and non-temporal (keep X resident in L2).
// ---------------------------------------------------------------------------
__global__ void bev_sample(const float* __restrict__ X,
                           const unsigned int* __restrict__ params,
                           float* __restrict__ out) {
    const int tile  = blockIdx.x;       // 79 tiles of 128 over q in [0,10000)
    const int h     = blockIdx.y;       // 5
    const int b     = blockIdx.z;       // 4
    const int t     = threadIdx.x;      // 256
    const int ql    = t & 127;
    const int chalf = t >> 7;           // 0 -> even channels, 1 -> odd channels
    const int q     = tile * 128 + ql;
    if (q >= QTOT) return;

    const int pidx = (b * H_RES + h) * QTOT + q;
    const uint4  offs = ((const uint4*) (params + (size_t)pidx * 8))[0];
    const float4 wts  = ((const float4*)(params + (size_t)pidx * 8))[1];

    const float* Xb = X + (size_t)b * CH * HW;
    float* ob = out + ((size_t)(b * CH) * H_RES + h) * QTOT + q;

#pragma unroll 4
    for (int c = chalf; c < CH; c += 2) {
        const float* plane = Xb + (size_t)c * HW;
        const float v = wts.x * plane[offs.x]
                      + wts.y * plane[offs.y]
                      + wts.z * plane[offs.z]
                      + wts.w * plane[offs.w];
        __builtin_nontemporal_store(v, ob + (size_t)c * (H_RES * QTOT));
    }
}

extern "C" void kernel_launch(void* const* d_in, const int* in_sizes, int n_in,
                              void* d_out, int out_size, void* d_ws, size_t ws_size,
                              hipStream_t stream) {
    const float* X     = (const float*)d_in[0];   // (4,256,96,320) f32
    const float* calib = (const float*)d_in[1];   // (4,3,3) f32
    float* out = (float*)d_out;                   // (4,1280,100,100) f32
    unsigned int* params = (unsigned int*)d_ws;   // 200000 * 32 B = 6.4 MB

    // Kernel 1: 12500 wave-chunks, 8 waves per 256-thread block
    bev_project_wmma<<<dim3((NCHUNKS + 7) / 8), dim3(256), 0, stream>>>(calib, params);

    // Kernel 2: 79 q-tiles x 5 h x 4 b
    bev_sample<<<dim3((QTOT + 127) / 128, H_RES, NB), dim3(256), 0, stream>>>(X, params, out);
}